// HyperLayer_50972671869236
// MI455X (gfx1250) — compile-verified
//
#include <hip/hip_runtime.h>

namespace {
constexpr int kB      = 16;
constexpr int kN      = 262144;
constexpr int kInDim  = 8192;
constexpr int kOutDim = 8192;
constexpr int kChunks = 32;    // chunks per batch row -> 512 workgroups
constexpr int kThreads = 256;
}

typedef unsigned int v4u __attribute__((ext_vector_type(4)));
typedef int          v4i __attribute__((ext_vector_type(4)));
typedef int          v8i __attribute__((ext_vector_type(8)));
typedef float        v4f __attribute__((ext_vector_type(4)));
typedef float        v2f __attribute__((ext_vector_type(2)));

__global__ __launch_bounds__(kThreads) void hyper_scatter(
    const float* __restrict__ x,
    const float* __restrict__ real_indices,   // [B,N,2] flat
    const float* __restrict__ real_values,    // [B,N]
    float* __restrict__ out) {
  __shared__ float x_lds[kInDim];   // 32 KB: gather table (one x row)
  __shared__ float acc[kOutDim];    // 32 KB: privatized output row

  const int b     = (int)blockIdx.x % kB;
  const int chunk = (int)blockIdx.x / kB;
  const int tid   = threadIdx.x;

  // Wave 0: issue one Tensor-Data-Mover DMA of x[b, :] (32 KB) into LDS.
#if __has_builtin(__builtin_amdgcn_tensor_load_to_lds)
  if (tid < 32) {
    unsigned long long ga =
        (unsigned long long)(const void*)(x + (size_t)b * kInDim);
    // Flat LDS addresses carry the DS byte offset in the low 32 bits.
    unsigned int lds_off = (unsigned int)(unsigned long long)(void*)x_lds;

    v4u g0;                                   // D# group 0 (ISA 8.3)
    g0.x = 1u;                                // count = 1 (valid descriptor)
    g0.y = lds_off;                           // lds_addr (bytes)
    g0.z = (unsigned int)ga;                  // global_addr[31:0]
    g0.w = (unsigned int)((ga >> 32) & 0x01FFFFFFull) | (2u << 30); // addr[56:32] | type=2

    v8i g1;                                   // D# group 1 (ISA 8.4)
    g1[0] = 2 << 16;                          // workgroup_mask=0, data_size=2 (4B)
    g1[1] = (int)((unsigned)kInDim << 16);    // tensor_dim0 = 8192 (lo16)
    g1[2] = 1 << 16;                          // tensor_dim0 hi=0, tensor_dim1 = 1
    g1[3] = (int)((unsigned)kInDim << 16);    // tensor_dim1 hi=0, tile_dim0 = 8192
    g1[4] = 1;                                // tile_dim1 = 1, tile_dim2 = 0
    g1[5] = kInDim;                           // tensor_dim0_stride = 8192
    g1[6] = 0;
    g1[7] = 0;
    v4i gz = {0, 0, 0, 0};                    // groups 2/3: unused dims = 0
#if __clang_major__ >= 23
    v8i gz8 = {0, 0, 0, 0, 0, 0, 0, 0};
    __builtin_amdgcn_tensor_load_to_lds(g0, g1, gz, gz, gz8, 0);
#else
    __builtin_amdgcn_tensor_load_to_lds(g0, g1, gz, gz, 0);
#endif
  }
#endif

  // All waves zero the privatized accumulator while the TDM streams x.
  for (int j = tid; j < kOutDim; j += kThreads) acc[j] = 0.0f;

#if __has_builtin(__builtin_amdgcn_tensor_load_to_lds)
  if (tid < 32) {
    __builtin_amdgcn_s_wait_tensorcnt(0);
  }
#else
  for (int j = tid; j < kInDim; j += kThreads)
    x_lds[j] = x[(size_t)b * kInDim + j];
#endif
  __syncthreads();

  constexpr int pairs  = kN / kChunks;        // 8192 samples per block
  constexpr int pairs2 = pairs / 2;           // process 2 samples / iter
  const size_t base = (size_t)b * kN + (size_t)chunk * pairs;
  const v4f* __restrict__ ri4 = (const v4f*)(real_indices + base * 2);
  const v2f* __restrict__ rv2 = (const v2f*)(real_values + base);

  for (int i = tid; i < pairs2; i += kThreads) {
    v4f p = __builtin_nontemporal_load(ri4 + i);  // global_load_b128 th:NT
    v2f v = __builtin_nontemporal_load(rv2 + i);  // global_load_b64  th:NT
#pragma unroll
    for (int s = 0; s < 2; ++s) {
      float i0 = (s == 0) ? p.x : p.z;
      float i1 = (s == 0) ? p.y : p.w;
      float vv = (s == 0) ? v.x : v.y;
      float f0 = floorf(i0), c0 = ceilf(i0);
      float f1 = floorf(i1), c1 = ceilf(i1);
      float a0 = 1.0f - (i0 - f0);            // 1-|f0 - i0|
      float b0 = 1.0f - (c0 - i0);            // 1-|c0 - i0|
      float a1 = 1.0f - (i1 - f1);
      float b1 = 1.0f - (c1 - i1);
      float xf = x_lds[(int)f1];              // LDS gathers
      float xc = x_lds[(int)c1];
      float t  = vv * fmaf(a1, xf, b1 * xc);  // shared factor of both rows
      atomicAdd(&acc[(int)f0], a0 * t);       // ds_add_f32 (no return)
      atomicAdd(&acc[(int)c0], b0 * t);       // ds_add_f32 (no return)
    }
  }
  __syncthreads();

  // Flush privatized row with fp32 global atomics (no return).
  float* orow = out + (size_t)b * kOutDim;
  for (int j = tid; j < kOutDim; j += kThreads) {
#if defined(__gfx1250__)
    unsafeAtomicAdd(&orow[j], acc[j]);        // global_atomic_add_f32
#else
    atomicAdd(&orow[j], acc[j]);
#endif
  }
}

__global__ void zero_f32(float* p, int n) {
  int i = blockIdx.x * blockDim.x + threadIdx.x;
  if (i < n) p[i] = 0.0f;
}

extern "C" void kernel_launch(void* const* d_in, const int* in_sizes, int n_in,
                              void* d_out, int out_size, void* d_ws, size_t ws_size,
                              hipStream_t stream) {
  (void)in_sizes; (void)n_in; (void)out_size; (void)d_ws; (void)ws_size;
  const float* x  = (const float*)d_in[0];
  const float* ri = (const float*)d_in[1];
  const float* rv = (const float*)d_in[2];
  float* out = (float*)d_out;

  const int total = kB * kOutDim;
  zero_f32<<<(total + 255) / 256, 256, 0, stream>>>(out, total);
  hyper_scatter<<<kB * kChunks, kThreads, 0, stream>>>(x, ri, rv, out);
}